// WavetableSynth_69191923138798
// MI455X (gfx1250) — compile-verified
//
#include <hip/hip_runtime.h>

// Wavetable synth for MI455X (gfx1250, wave32).
//
// Pipeline (all on `stream`):
//   k_block_reduce : per-2048-element block sums of increments (float2 loads)
//   k_wt_mean      : 512 column means of the 64x512 wavetable bank
//   k_scan_blocks  : exclusive scan of block sums (single block, 1024 thr)
//   k_scan_synth   : per-wave 256-element inclusive scan done with
//                    V_WMMA_F32_16X16X4_F32 (tensor-core scan), then
//                    index -> lerp from LDS-resident column means.
//
// Scan-as-matmul: X (16x16 row-major tile of increments),
//   R = X * U            (U upper-tri ones)  -> row-inclusive prefix
//   D = Lstrict * B + R  (B[k][n]=rowsum[k]) -> + sum of earlier rows
// f32 WMMA keeps full cumsum precision (values reach ~7e6).
//
// Tail handling: load addresses are CLAMPED (min with last valid pair) rather
// than zero-filled. For a prefix scan this is safe: clamped garbage only
// affects scan positions >= the clamped elements, all of which have idx >= T
// and are never stored (and only feed wave/row offsets of equally-dead
// positions). This keeps every load unconditional -> batched b64 loads.
//
// Cross-wave carry: lanes 15/31 deposit per-half rowsum partials into
// sh_part[16]; every thread then derives its wave's exclusive offset from 16
// LDS broadcasts. No serial single-thread scan, only ONE block barrier.

typedef __attribute__((ext_vector_type(2))) float v2f;
typedef __attribute__((ext_vector_type(8))) float v8f;

#define WT_LEN      512
#define TILE        256    // elements per wave (16x16 f32 tile)
#define BLOCK_ELEMS 2048   // 8 waves * 256 elements

__device__ __forceinline__ int imin(int a, int b) { return a < b ? a : b; }

// ---------------- phase 1: block sums ----------------
__global__ void __launch_bounds__(256)
k_block_reduce(const float* __restrict__ pitch, const int* __restrict__ srp,
               int T, float* __restrict__ blockSums) {
  __shared__ float red[256];
  const float scale = (float)WT_LEN / (float)srp[0];
  const int t = threadIdx.x;
  const int b = blockIdx.x;
  const float2* __restrict__ p2 = (const float2*)pitch;
  const int pbase = b * (BLOCK_ELEMS / 2);
  float s = 0.0f;
  if ((b + 1) * BLOCK_ELEMS <= T) {        // full block: no guards
#pragma unroll
    for (int i = 0; i < 4; ++i) {
      const float2 v = p2[pbase + i * 256 + t];
      s += v.x + v.y;
    }
  } else {                                  // tail block only
    const int np = T >> 1;                  // T is even
#pragma unroll
    for (int i = 0; i < 4; ++i) {
      const int pi = pbase + i * 256 + t;
      if (pi < np) { const float2 v = p2[pi]; s += v.x + v.y; }
    }
  }
  red[t] = s * scale;
  __syncthreads();
  for (int off = 128; off > 0; off >>= 1) {
    if (t < off) red[t] += red[t + off];
    __syncthreads();
  }
  if (t == 0) blockSums[b] = red[0];
}

// ---------------- phase 1b: wavetable column means ----------------
__global__ void __launch_bounds__(256)
k_wt_mean(const float* __restrict__ wt, float* __restrict__ wtMean) {
  const int c = blockIdx.x * 256 + threadIdx.x;  // 0..511
  float s = 0.0f;
#pragma unroll
  for (int r = 0; r < 64; ++r) s += wt[r * WT_LEN + c];
  wtMean[c] = s * (1.0f / 64.0f);
}

// ---------------- phase 2: exclusive scan of block sums ----------------
// nB = ceil(1.32M / 2048) = 646 <= 1024, single block Hillis-Steele.
__global__ void __launch_bounds__(1024)
k_scan_blocks(const float* __restrict__ blockSums,
              float* __restrict__ blockOff, int nB) {
  __shared__ float buf[2][1024];
  const int t = threadIdx.x;
  buf[0][t] = (t < nB) ? blockSums[t] : 0.0f;
  __syncthreads();
  int src = 0;
#pragma unroll
  for (int off = 1; off < 1024; off <<= 1) {
    float x = buf[src][t];
    if (t >= off) x += buf[src][t - off];
    buf[src ^ 1][t] = x;
    src ^= 1;
    __syncthreads();
  }
  if (t < nB) blockOff[t] = (t == 0) ? 0.0f : buf[src][t - 1];
}

// ---------------- phase 3: WMMA scan + synthesis ----------------
__global__ void __launch_bounds__(256)
k_scan_synth(const float* __restrict__ pitch,
             const float* __restrict__ blockOff,
             const float* __restrict__ wtMean,
             const int* __restrict__ srp, int T,
             float* __restrict__ out) {
  __shared__ float sh_mean[WT_LEN];      // 2 KB table of column means
  __shared__ float sh_rowsum[8 * 16];    // 16 rowsums per wave
  __shared__ float sh_part[16];          // per-half-wave rowsum partials

  const int tid  = threadIdx.x;
  const int wave = tid >> 5;
  const int lane = tid & 31;
  const int col  = lane & 15;  // A-row index m / C-D column N
  const int half = lane >> 4;  // lane-half -> K/M high selector

  sh_mean[tid]       = wtMean[tid];
  sh_mean[tid + 256] = wtMean[tid + 256];

  const float scale = (float)WT_LEN / (float)srp[0];
  const int base = blockIdx.x * BLOCK_ELEMS + wave * TILE;

  // ---- A-slice loads: 4 independent clamped b64 loads (full MLP) ----
  // element e = base + 16*col + 4*kk + 2*half is always even ->
  // float2 pair index = base/2 + 8*col + 2*kk + half.
  const float2* __restrict__ p2 = (const float2*)pitch;
  const int vbase = (base >> 1) + 8 * col + half;
  const int vmax  = (T >> 1) - 1;        // clamp target (tail-safe, see top)
  float2 x[4];
#pragma unroll
  for (int kk = 0; kk < 4; ++kk) x[kk] = p2[imin(vbase + 2 * kk, vmax)];

  // ---- R = X * U : row-inclusive prefix of the 16x16 tile ----
  // A layout (ISA 7.12.2, 32-bit A 16x4): lanes 0-15 hold K=k0,k0+1,
  // lanes 16-31 hold K=k0+2,k0+3, row M = lane&15.
  v8f r = {};
#pragma unroll
  for (int kk = 0; kk < 4; ++kk) {
    const int k0 = 4 * kk + 2 * half;
    v2f a, b;
    a.x = x[kk].x * scale;
    a.y = x[kk].y * scale;
    // B = U slice: U[k][n] = (n >= k), n = col
    b.x = (col >= k0)     ? 1.0f : 0.0f;
    b.y = (col >= k0 + 1) ? 1.0f : 0.0f;
    r = __builtin_amdgcn_wmma_f32_16x16x4_f32(false, a, false, b,
                                              (short)0, r, false, false);
  }

  // Rowsums live in column 15 of R: lane15 has rows 0-7, lane31 rows 8-15.
  // Also deposit each half-wave's 8-row partial sum for the cross-wave carry.
  if (col == 15) {
#pragma unroll
    for (int p = 0; p < 8; ++p)
      sh_rowsum[wave * 16 + half * 8 + p] = r[p];
    const float ps = ((r[0] + r[1]) + (r[2] + r[3])) +
                     ((r[4] + r[5]) + (r[6] + r[7]));
    sh_part[wave * 2 + half] = ps;
  }
  __syncthreads();   // single block barrier

  // Every thread derives its wave's exclusive offset from 16 broadcasts.
  float parts[16];
#pragma unroll
  for (int i = 0; i < 16; ++i) parts[i] = sh_part[i];  // -> 4x ds_load_b128
  const int lim = 2 * wave;
  float woff = 0.0f;
#pragma unroll
  for (int i = 0; i < 16; ++i) woff += (i < lim) ? parts[i] : 0.0f;

  // ---- D = Lstrict * Bcast(rowsum) + R : add prefix of earlier rows ----
  v8f d = r;
#pragma unroll
  for (int kk = 0; kk < 4; ++kk) {
    const int k0 = 4 * kk + 2 * half;
    v2f a, b;
    a.x = (k0     < col) ? 1.0f : 0.0f;   // Lstrict[m][k] = (k < m)
    a.y = (k0 + 1 < col) ? 1.0f : 0.0f;
    b.x = sh_rowsum[wave * 16 + k0];       // B[k][n] = rowsum[k] (bcast)
    b.y = sh_rowsum[wave * 16 + k0 + 1];
    d = __builtin_amdgcn_wmma_f32_16x16x4_f32(false, a, false, b,
                                              (short)0, d, false, false);
  }

  // carry = global exclusive offset; subtract inc[0] per the reference.
  const float inc0  = pitch[0] * scale;
  const float carry = blockOff[blockIdx.x] + woff - inc0;

  // ---- finalize: mod, lerp from LDS means, store ----
  float res[8];
#pragma unroll
  for (int p = 0; p < 8; ++p) {
    float v = d[p] + carry;
    v = fmodf(v, (float)WT_LEN);
    v = (v < 0.0f) ? v + (float)WT_LEN : v;
    const float fl    = floorf(v);
    const float alpha = v - fl;
    const int lo = ((int)fl)       & (WT_LEN - 1);
    const int hi = ((int)ceilf(v)) & (WT_LEN - 1);   // ceil(index) mod L
    const float mlo = sh_mean[lo];
    const float mhi = sh_mean[hi];
    res[p] = mlo + alpha * (mhi - mlo);
  }

  // idx = base + 16*(p + 8*half) + col
  const int sbase = base + 128 * half + col;
  if (base + TILE <= T) {                 // wave-uniform fast path
#pragma unroll
    for (int p = 0; p < 8; ++p) out[sbase + 16 * p] = res[p];
  } else {                                // tail wave(s) only
#pragma unroll
    for (int p = 0; p < 8; ++p) {
      const int idx = sbase + 16 * p;
      if (idx < T) out[idx] = res[p];
    }
  }
}

extern "C" void kernel_launch(void* const* d_in, const int* in_sizes, int n_in,
                              void* d_out, int out_size, void* d_ws,
                              size_t ws_size, hipStream_t stream) {
  const float* pitch = (const float*)d_in[0];
  // d_in[1] = amplitude: unused by the reference output
  const float* wt    = (const float*)d_in[2];
  const int*   srp   = (const int*)d_in[3];
  // d_in[4] = duration_secs: implied by T = in_sizes[0]

  const int T  = in_sizes[0];
  const int nB = (T + BLOCK_ELEMS - 1) / BLOCK_ELEMS;  // 646 for T=1.323M

  float* ws        = (float*)d_ws;
  float* blockSums = ws;            // nB floats
  float* blockOff  = ws + nB;       // nB floats
  float* wtMean    = ws + 2 * nB;   // 512 floats  (~7.2 KB total)

  k_block_reduce<<<nB, 256, 0, stream>>>(pitch, srp, T, blockSums);
  k_wt_mean<<<2, 256, 0, stream>>>(wt, wtMean);
  k_scan_blocks<<<1, 1024, 0, stream>>>(blockSums, blockOff, nB);
  k_scan_synth<<<nB, 256, 0, stream>>>(pitch, blockOff, wtMean, srp, T,
                                       (float*)d_out);
}